// transfer_matrix_60936995995990
// MI455X (gfx1250) — compile-verified
//
#include <hip/hip_runtime.h>
#include <math.h>

// Fresnel transfer-matrix (s-pol), 61 angles x 16384 wavelengths, 7 interfaces.
// Physics in f64 (reference is complex128); angle-axis mean/var reduced with
// V_WMMA_F32_16X16X4_F32 (full-f32 matrix op -> no precision loss vs f32 out).

typedef __attribute__((ext_vector_type(2))) float v2f;
typedef __attribute__((ext_vector_type(8))) float v8f;

#define NMAT   8
#define NANG   61
#define ASLOTS 64     // 61 angles padded to 64 (multiple of K=4 chunks)
#define WPB    16     // wavelengths per block (= WMMA N)
#define TPB    256
#define PI_D   3.14159265358979323846

struct cd { double re, im; };

__device__ __forceinline__ cd cmul(cd a, cd b) {
    return { a.re * b.re - a.im * b.im, a.re * b.im + a.im * b.re };
}
__device__ __forceinline__ cd cadd(cd a, cd b) { return { a.re + b.re, a.im + b.im }; }
__device__ __forceinline__ cd csub(cd a, cd b) { return { a.re - b.re, a.im - b.im }; }
__device__ __forceinline__ cd cdiv(cd a, cd b) {
    double inv = 1.0 / (b.re * b.re + b.im * b.im);
    return { (a.re * b.re + a.im * b.im) * inv, (a.im * b.re - a.re * b.im) * inv };
}
// principal branch complex sqrt (matches cos(arcsin(z)) for principal arcsin)
__device__ __forceinline__ cd csqrt_p(cd z) {
    double r  = sqrt(z.re * z.re + z.im * z.im);
    double re = sqrt(fmax(0.5 * (r + z.re), 0.0));
    double im = copysign(sqrt(fmax(0.5 * (r - z.re), 0.0)), z.im);
    return { re, im };
}

__global__ __launch_bounds__(TPB)
void transfer_matrix_kernel(const float* __restrict__ x,
                            const float* __restrict__ n0,
                            const float* __restrict__ k0,
                            const float* __restrict__ wl,
                            float* __restrict__ out,
                            int N)
{
    __shared__ float tile[ASLOTS][WPB];   // [angle slot][wavelength-in-block]
    __shared__ float meanLds[WPB];

    const int tid   = threadIdx.x;
    const int wloc  = tid & 15;           // wavelength column 0..15
    const int abase = tid >> 4;           // 0..15 ; angles abase + {0,16,32,48}
    const int wglob = blockIdx.x * WPB + wloc;
    const bool valid = (wglob < N);

    // zero the LDS tile (pad rows 61..63 must stay zero)
    {
        float* t = &tile[0][0];
        for (int i = tid; i < ASLOTS * WPB; i += TPB) t[i] = 0.0f;
    }
    __syncthreads();

    // per-thread (one wavelength) material data
    double nre[NMAT], nim[NMAT], xv[NMAT - 2];
    double wlv = 500.0, n00 = 1.0;
    if (valid) {
        for (int m = 0; m < NMAT; ++m) {
            nre[m] = (double)n0[wglob * NMAT + m];
            nim[m] = (double)k0[wglob * NMAT + m];
        }
        wlv = (double)wl[wglob];
        n00 = nre[0];
    }
    for (int m = 0; m < NMAT - 2; ++m) xv[m] = (double)x[m];

    // ---- f64 complex recurrence: 4 angles per thread ----
    float Tval[4];
    for (int j = 0; j < 4; ++j) {
        int a = abase + j * 16;
        Tval[j] = 0.0f;
        if (valid && a < NANG) {
            double rad = PI_D * (double)a / 180.0;
            double sa, ca;
            sincos(rad, &sa, &ca);
            cd s   = { sa, 0.0 };           // sin(theta)
            cd c   = { ca, 0.0 };           // cos(theta)
            cd m00 = { 1.0, 0.0 };          // row 0 of smat is all we need
            cd m01 = { 0.0, 0.0 };
            cd njct = { 0.0, 0.0 };
            #pragma unroll
            for (int i = 0; i < NMAT - 1; ++i) {
                cd ni = { nre[i],     nim[i]     };
                cd nj = { nre[i + 1], nim[i + 1] };
                cd s2 = cdiv(cmul(s, ni), nj);                 // sin(theta_t)
                cd om = { 1.0 - (s2.re * s2.re - s2.im * s2.im),
                          -2.0 * s2.re * s2.im };              // 1 - s2^2
                cd ct = csqrt_p(om);                           // cos(theta_t)
                cd nici = cmul(ni, c);
                njct    = cmul(nj, ct);
                cd denom = cadd(nici, njct);
                cd r  = cdiv(csub(nici, njct), denom);
                cd t2 = { 2.0 * nici.re, 2.0 * nici.im };
                cd invt = cdiv(denom, t2);                     // 1/t = denom/(2 ni ci)
                cd A = cadd(m00, cmul(m01, r));                // (m00 + m01 r)
                cd B = cadd(cmul(m00, r), m01);                // (m00 r + m01)
                m00 = cmul(A, invt);
                m01 = cmul(B, invt);
                if (i < NMAT - 2) {
                    double kph = 2.0 * PI_D * xv[i] / wlv;
                    double phr = kph * njct.re;
                    double phi = kph * njct.im;
                    double sp, cp;
                    sincos(phr, &sp, &cp);
                    double em = exp(phi);
                    double ep = 1.0 / em;
                    cd eM = { em * cp, -em * sp };             // e^{-i ph}
                    cd eP = { ep * cp,  ep * sp };             // e^{+i ph}
                    m00 = cmul(m00, eM);                       // column 0 scale
                    m01 = cmul(m01, eP);                       // column 1 scale
                }
                s = s2; c = ct;
            }
            double mag2 = m00.re * m00.re + m00.im * m00.im;   // |smat00|^2
            double T = njct.re / (mag2 * n00 * ca);            // |1/m00|^2 Re(nj ct)/(n0 cos)
            Tval[j] = (float)T;
            tile[a][wloc] = Tval[j];
        }
    }
    __syncthreads();

    // ---- pass 1: column sums over 64 angle rows via WMMA f32 16x16x4 ----
    // A = all-ones (layout-invariant); B column n lives in lanes {n, n+16};
    // sum over K is order-invariant, and every D row equals the colsums.
    const int lane = tid & 31;
    const int ncol = lane & 15;
    const int half = (lane >> 4) & 1;
    v2f aone = { 1.0f, 1.0f };

    v8f acc = {};
    #pragma unroll
    for (int cb = 0; cb < 16; ++cb) {
        int r0 = cb * 4 + half * 2;
        v2f b;
        b[0] = tile[r0 + 0][ncol];
        b[1] = tile[r0 + 1][ncol];
        acc = __builtin_amdgcn_wmma_f32_16x16x4_f32(
            false, aone, false, b, (short)0, acc, false, false);
    }
    float meanv = acc[0] * (1.0f / (float)NANG);
    if (tid < 16) meanLds[tid] = meanv;        // wave 0, ncol == tid
    __syncthreads();

    // ---- pass 2: squared deviations, then WMMA colsum again -> variance ----
    float mw = meanLds[wloc];
    for (int j = 0; j < 4; ++j) {
        int a = abase + j * 16;
        if (valid && a < NANG) {
            float d = Tval[j] - mw;
            tile[a][wloc] = d * d;             // pad rows stay 0
        }
    }
    __syncthreads();

    v8f acc2 = {};
    #pragma unroll
    for (int cb = 0; cb < 16; ++cb) {
        int r0 = cb * 4 + half * 2;
        v2f b;
        b[0] = tile[r0 + 0][ncol];
        b[1] = tile[r0 + 1][ncol];
        acc2 = __builtin_amdgcn_wmma_f32_16x16x4_f32(
            false, aone, false, b, (short)0, acc2, false, false);
    }
    float varv = acc2[0] * (1.0f / (float)(NANG - 1));   // ddof=1

    if (tid < 16) {
        int w = blockIdx.x * WPB + tid;
        if (w < N) {
            out[w]     = meanv;
            out[N + w] = varv;
        }
    }
}

extern "C" void kernel_launch(void* const* d_in, const int* in_sizes, int n_in,
                              void* d_out, int out_size, void* d_ws, size_t ws_size,
                              hipStream_t stream) {
    const float* x   = (const float*)d_in[0];   // [6]
    const float* n0v = (const float*)d_in[1];   // [N,8]
    const float* k0v = (const float*)d_in[2];   // [N,8]
    const float* wlv = (const float*)d_in[3];   // [N]
    float* out = (float*)d_out;                 // [2N] = mean ++ var
    int N = in_sizes[3];

    int grid = (N + WPB - 1) / WPB;
    transfer_matrix_kernel<<<grid, TPB, 0, stream>>>(x, n0v, k0v, wlv, out, N);
}